// Transform_Camera_67954972557734
// MI455X (gfx1250) — compile-verified
//
#include <hip/hip_runtime.h>
#include <hip/hip_bf16.h>
#include <stdint.h>

// Problem constants (from reference setup_inputs)
#define DIFF   16
#define NB     8
#define NC     3
#define NT     32
#define NH     224
#define NW     224
#define TILE_H 16                 // output rows per block; needs <=16 source rows (scale 14/15)

#define SCL  (224.0f / 240.0f)    // 14/15
#define OFF  (-1.0f / 30.0f)      // 0.5*14/15 - 0.5

#if defined(__gfx1250__) && __has_builtin(__builtin_amdgcn_tensor_load_to_lds)
#define USE_TDM 1
#endif

#ifdef USE_TDM
typedef unsigned int v4u __attribute__((ext_vector_type(4)));
typedef int          v4i __attribute__((ext_vector_type(4)));
typedef int          v8i __attribute__((ext_vector_type(8)));

// Issue one 2-D TDM load: tile_h rows x tile_w f32 elements, row stride = stride
// elements, from gptr into LDS byte offset lds_off. Tensor dims == tile dims so
// no OOB path is exercised. workgroup_mask = 0 (not in a cluster).
__device__ inline void tdm_load_tile_f32(uint32_t lds_off, const float* gptr,
                                         uint32_t tile_w, uint32_t tile_h,
                                         uint32_t stride) {
  uint64_t ga = (uint64_t)(uintptr_t)gptr;
  v4u g0;
  g0.x = 1u;                                   // count=1, user descriptor
  g0.y = lds_off;                              // lds_addr (bytes)
  g0.z = (uint32_t)(ga & 0xFFFFFFFFu);         // global_addr[31:0]
  g0.w = (uint32_t)((ga >> 32) & 0x01FFFFFFu)  // global_addr[56:32]
       | (2u << 30);                           // type = 2 ("image")
  v8i g1;
  g1[0] = (int)(2u << 16);                           // data_size=2 (4B); mask=0
  g1[1] = (int)((tile_w & 0xFFFFu) << 16);           // tensor_dim0[15:0]
  g1[2] = (int)((tile_w >> 16) | ((tile_h & 0xFFFFu) << 16));   // dim0 hi | dim1 lo
  g1[3] = (int)(((tile_h >> 16) & 0xFFFFu) | ((tile_w & 0xFFFFu) << 16)); // dim1 hi | tile_dim0
  g1[4] = (int)(tile_h & 0xFFFFu);                   // tile_dim1 | tile_dim2=0
  g1[5] = (int)stride;                               // tensor_dim0_stride[31:0]
  g1[6] = 0;                                         // stride hi | dim1_stride lo
  g1[7] = 0;
  v4i z4 = {0, 0, 0, 0};
#if __has_include(<hip/amd_detail/amd_gfx1250_TDM.h>)
  v8i z8 = {0, 0, 0, 0, 0, 0, 0, 0};
  __builtin_amdgcn_tensor_load_to_lds(g0, g1, z4, z4, z8, 0);   // clang-23 form
#else
  __builtin_amdgcn_tensor_load_to_lds(g0, g1, z4, z4, 0);       // ROCm 7.2 form
#endif
}
#endif // USE_TDM

__global__ __launch_bounds__(256)
void shake_crop_kernel(const float* __restrict__ video,
                       const int* __restrict__ shake_h,
                       const int* __restrict__ shake_w,
                       float* __restrict__ out) {
  __shared__ float tile[TILE_H * NW];   // 16 source rows (14.3 KB)
  __shared__ uint2 xlut[NW];            // {x0 | x1<<16, bits(fx)}
  __shared__ uint2 ylut[TILE_H];        // {ry0 | ry1<<16, bits(fy)}

  const int t   = blockIdx.x;           // frame
  const int bc  = blockIdx.y;           // fused B*C
  const int h0  = blockIdx.z * TILE_H;  // output row tile
  const int tid = threadIdx.x;

  const int sh = shake_h[t];
  const int sw = shake_w[t];

  // First source row of the tile. 15 output rows span exactly 14 source rows,
  // so (with both bilinear taps and edge clamping) 16 consecutive source rows
  // starting at y_start cover the whole tile.
  const float sy0 = (float)(h0 + sh) * SCL + OFF;
  int y_start = (int)floorf(sy0);
  y_start = min(max(y_start, 0), NH - TILE_H);

  const float* frame = video + (size_t)(bc * NT + t) * (size_t)(NH * NW);

#ifdef USE_TDM
  // Wave-uniform scalar guard: TDM ops IGNORE EXEC, so an EXEC-predicated
  // region would issue the DMA once per wave (8x duplicate traffic). Forcing
  // the predicate through readfirstlane puts it in an SGPR -> s_cbranch, so
  // only wave 0's instruction stream contains tensor_load_to_lds.
  const int wave_id = __builtin_amdgcn_readfirstlane((int)(tid >> 5));
  if (wave_id == 0) {
    tdm_load_tile_f32((uint32_t)(uintptr_t)tile,
                      frame + (size_t)y_start * NW, NW, TILE_H, NW);
  }
#else
  for (int i = tid; i < TILE_H * NW; i += 256) {
    const int r = i / NW;
    tile[i] = frame[(size_t)(y_start + r) * NW + (i - r * NW)];
  }
#endif

  // Coordinate LUTs (overlap with the in-flight TDM transfer).
  if (tid < NW) {
    const float sx = (float)(tid + sw) * SCL + OFF;
    const float xf = floorf(sx);
    const int   ix = (int)xf;
    const int   x0 = max(ix, 0);
    const int   x1 = min(ix + 1, NW - 1);
    xlut[tid] = make_uint2((uint32_t)x0 | ((uint32_t)x1 << 16),
                           __float_as_uint(sx - xf));
  }
  if (tid < TILE_H) {
    const float sy = (float)(h0 + tid + sh) * SCL + OFF;
    const float yf = floorf(sy);
    const int   iy = (int)yf;
    const int   r0 = max(iy, 0) - y_start;            // in [0, TILE_H-1]
    const int   r1 = min(iy + 1, NH - 1) - y_start;   // in [0, TILE_H-1]
    ylut[tid] = make_uint2((uint32_t)r0 | ((uint32_t)r1 << 16),
                           __float_as_uint(sy - yf));
  }

#ifdef USE_TDM
  if (wave_id == 0) __builtin_amdgcn_s_wait_tensorcnt(0);  // wave 0 owns TENSORcnt
#endif
  __syncthreads();

  float* optr = out + ((size_t)(bc * NT + t) * NH + h0) * (size_t)NW;

#pragma unroll
  for (int i = 0; i < (TILE_H * NW) / 256; ++i) {      // 14 pixels / thread
    const int idx = i * 256 + tid;
    const int hh  = idx / NW;
    const int w   = idx - hh * NW;

    const uint2 xc = xlut[w];
    const uint2 yc = ylut[hh];
    const int x0 = (int)(xc.x & 0xFFFFu), x1 = (int)(xc.x >> 16);
    const int r0 = (int)(yc.x & 0xFFFFu), r1 = (int)(yc.x >> 16);
    const float fx = __uint_as_float(xc.y);
    const float fy = __uint_as_float(yc.y);

    const float* row0 = &tile[r0 * NW];
    const float* row1 = &tile[r1 * NW];
    const float a00 = row0[x0], a01 = row0[x1];
    const float a10 = row1[x0], a11 = row1[x1];

    const float top = a00 + fx * (a01 - a00);
    const float bot = a10 + fx * (a11 - a10);
    float v = top + fy * (bot - top);
    v = fminf(fmaxf(v, 0.0f), 1.0f);                   // reference clip

    optr[(size_t)hh * NW + w] = v;
  }
}

extern "C" void kernel_launch(void* const* d_in, const int* in_sizes, int n_in,
                              void* d_out, int out_size, void* d_ws, size_t ws_size,
                              hipStream_t stream) {
  (void)in_sizes; (void)n_in; (void)out_size; (void)d_ws; (void)ws_size;
  const float* video = (const float*)d_in[0];
  const int*   sh    = (const int*)d_in[1];
  const int*   sw    = (const int*)d_in[2];
  float*       out   = (float*)d_out;

  dim3 grid(NT, NB * NC, NH / TILE_H);   // (32, 24, 14)
  shake_crop_kernel<<<grid, 256, 0, stream>>>(video, sh, sw, out);
}